// Code_Wise_Attention_6158983102794
// MI455X (gfx1250) — compile-verified
//
#include <hip/hip_runtime.h>

// Code-Wise Attention for MI455X (gfx1250, wave32).
// Phase 1: fp32 WMMA (v_wmma_f32_16x16x4_f32) GEMM with fused row-max,
//          B-panel double-buffered in LDS via async global->LDS copies
//          (GLOBAL_LOAD_ASYNC_TO_LDS_B128, ASYNCcnt) when available.
// Phase 2: softmax over context axis + weighted context sum.

typedef __attribute__((ext_vector_type(2))) float v2f;
typedef __attribute__((ext_vector_type(8))) float v8f;
typedef __attribute__((ext_vector_type(4))) int   v4i;

#define BATCH   8
#define LQ      2048
#define LC      4096
#define DIM     256
#define M_TILE  128        // lc rows per block (8 waves x 16)
#define N_CHUNK 64         // lq cols per LDS stage (4 WMMA N-tiles)
#define NCHUNKS (LQ / N_CHUNK)
#define LDP     (DIM + 4)  // row pitch: 260 floats = 1040B -> 16B-aligned rows,
                           // bank stride 4 -> conflict-free 16-lane column reads
#define B_BUF   (N_CHUNK * LDP)

#if __has_builtin(__builtin_amdgcn_global_load_async_to_lds_b128) && \
    __has_builtin(__builtin_amdgcn_s_wait_asynccnt)
#define USE_ASYNC_LDS 1
#define GAS __attribute__((address_space(1)))
#define LAS __attribute__((address_space(3)))
#else
#define USE_ASYNC_LDS 0
#endif

__global__ __launch_bounds__(256) void
scores_wmma_kernel(const float* __restrict__ query,
                   const float* __restrict__ context,
                   float* __restrict__ scores)
{
    extern __shared__ float smem[];
    float* As  = smem;                    // [M_TILE][LDP]
    float* Bs0 = smem + M_TILE * LDP;     // 2 x [N_CHUNK][LDP] (ping/pong)

    const int tid  = threadIdx.x;
    const int wid  = tid >> 5;            // wave id 0..7 -> M sub-tile
    const int lane = tid & 31;
    const int m    = lane & 15;           // row-within-tile lane index
    const int hk   = lane >> 4;           // half selector (K pair select)
    const int b    = blockIdx.y;
    const int lc0  = blockIdx.x * M_TILE;

    const float4* ctx4 = reinterpret_cast<const float4*>(
        context + ((size_t)b * LC + lc0) * DIM);
    const float4* qry4 = reinterpret_cast<const float4*>(
        query + (size_t)b * LQ * DIM);

#if USE_ASYNC_LDS
    // Issue 16 async B128 copies per thread: query chunk c -> LDS buffer `buf`.
    auto issueB = [&](int c, int buf) {
        const float4* src = qry4 + (size_t)c * N_CHUNK * (DIM / 4);
        float* dstbase = Bs0 + buf * B_BUF;
        #pragma unroll
        for (int i = 0; i < 16; ++i) {           // 4096 float4 / 256 threads
            int f4   = tid + i * 256;
            int row  = f4 >> 6;                  // 64 float4 per row
            int col4 = f4 & 63;
            __builtin_amdgcn_global_load_async_to_lds_b128(
                (GAS v4i*)(src + row * 64 + col4),
                (LAS v4i*)(dstbase + row * LDP + col4 * 4),
                0, 0);
        }
    };
    issueB(0, 0);   // prefetch chunk 0 while we stage A below
#else
    auto stageB = [&](int c, int buf) {
        const float4* src = qry4 + (size_t)c * N_CHUNK * (DIM / 4);
        float* dstbase = Bs0 + buf * B_BUF;
        #pragma unroll
        for (int i = 0; i < 16; ++i) {
            int f4   = tid + i * 256;
            int row  = f4 >> 6;
            int col4 = f4 & 63;
            float4 v = src[row * 64 + col4];
            float* dst = dstbase + row * LDP + col4 * 4;
            dst[0] = v.x; dst[1] = v.y; dst[2] = v.z; dst[3] = v.w;
        }
    };
#endif

    // ---- Stage A panel: 128 x 256 fp32 (context rows), once per block ----
    #pragma unroll
    for (int i = 0; i < 32; ++i) {               // 8192 float4 / 256 threads
        int f4   = tid + i * 256;
        int row  = f4 >> 6;
        int col4 = f4 & 63;
        float4 v = ctx4[row * 64 + col4];
        float* dst = As + row * LDP + col4 * 4;
        dst[0] = v.x; dst[1] = v.y; dst[2] = v.z; dst[3] = v.w;
    }

    const float* Arow = As + (wid * 16 + m) * LDP + 2 * hk;

    v8f vmax;
    #pragma unroll
    for (int r = 0; r < 8; ++r) vmax[r] = -__builtin_inff();

    __syncthreads();   // A panel ready (async B0 still in flight)

    for (int chunk = 0; chunk < NCHUNKS; ++chunk) {
        const int cur = chunk & 1;

#if USE_ASYNC_LDS
        if (chunk + 1 < NCHUNKS) {
            issueB(chunk + 1, cur ^ 1);           // prefetch next into other buf
            __builtin_amdgcn_s_wait_asynccnt(16); // current chunk's 16 done
        } else {
            __builtin_amdgcn_s_wait_asynccnt(0);
        }
        __syncthreads();   // all waves' async writes to buf[cur] visible
#else
        stageB(chunk, cur);
        __syncthreads();
#endif

        const float* Bbase = Bs0 + cur * B_BUF + m * LDP + 2 * hk;
        const float* Brow0 = Bbase;
        const float* Brow1 = Bbase + 16 * LDP;
        const float* Brow2 = Bbase + 32 * LDP;
        const float* Brow3 = Bbase + 48 * LDP;

        v8f c0 = {}, c1 = {}, c2 = {}, c3 = {};
        #pragma unroll 8
        for (int k = 0; k < DIM; k += 4) {
            v2f a  = *(const v2f*)(Arow  + k);
            v2f b0 = *(const v2f*)(Brow0 + k);
            v2f b1 = *(const v2f*)(Brow1 + k);
            v2f b2 = *(const v2f*)(Brow2 + k);
            v2f b3 = *(const v2f*)(Brow3 + k);
            c0 = __builtin_amdgcn_wmma_f32_16x16x4_f32(false, a, false, b0,
                                                       (short)0, c0, false, false);
            c1 = __builtin_amdgcn_wmma_f32_16x16x4_f32(false, a, false, b1,
                                                       (short)0, c1, false, false);
            c2 = __builtin_amdgcn_wmma_f32_16x16x4_f32(false, a, false, b2,
                                                       (short)0, c2, false, false);
            c3 = __builtin_amdgcn_wmma_f32_16x16x4_f32(false, a, false, b3,
                                                       (short)0, c3, false, false);
        }
        // Fold this chunk's 4 tiles into the running elementwise max:
        // each (VGPR r, lane) is a fixed (M,N) slot -> max over lq is legal.
        #pragma unroll
        for (int r = 0; r < 8; ++r) {
            float v = fmaxf(fmaxf(c0[r], c1[r]), fmaxf(c2[r], c3[r]));
            vmax[r] = fmaxf(vmax[r], v);
        }
        __syncthreads();   // all waves done reading buf[cur] before refill
    }

    // Cross-lane max within each 16-lane half (N runs over lanes):
    // after this, register r in half h holds rowmax of row M = r + 8*h.
    #pragma unroll
    for (int r = 0; r < 8; ++r) {
        float v = vmax[r];
        v = fmaxf(v, __shfl_xor(v, 1, 32));
        v = fmaxf(v, __shfl_xor(v, 2, 32));
        v = fmaxf(v, __shfl_xor(v, 4, 32));
        v = fmaxf(v, __shfl_xor(v, 8, 32));
        vmax[r] = v;
    }
    if (m == 0) {   // lane 0 writes rows 0..7, lane 16 writes rows 8..15
        float* srow = scores + (size_t)b * LC + lc0 + wid * 16 + 8 * hk;
        #pragma unroll
        for (int r = 0; r < 8; ++r) srow[r] = vmax[r];
    }
}

// ---- Phase 2: softmax over Lc, weighted context sum. One block per batch. ----
__global__ __launch_bounds__(256) void
softmax_ctxsum_kernel(const float* __restrict__ context,
                      const float* __restrict__ scores,
                      float* __restrict__ out)
{
    __shared__ float s_s[LC];     // 16 KB
    __shared__ float red[256];

    const int tid = threadIdx.x;
    const int b   = blockIdx.x;
    const float* sc = scores + (size_t)b * LC;

    float lm = -__builtin_inff();
    for (int i = tid; i < LC; i += 256) {
        float v = sc[i];
        s_s[i] = v;
        lm = fmaxf(lm, v);
    }
    red[tid] = lm;
    __syncthreads();
    for (int s = 128; s > 0; s >>= 1) {
        if (tid < s) red[tid] = fmaxf(red[tid], red[tid + s]);
        __syncthreads();
    }
    const float mx = red[0];

    const float* ctx = context + (size_t)b * LC * DIM;
    float acc = 0.f, sumw = 0.f;
    for (int lc = 0; lc < LC; ++lc) {
        float w = expf(s_s[lc] - mx);   // same w computed by all 256 lanes
        sumw += w;
        acc  += w * ctx[(size_t)lc * DIM + tid];   // coalesced: thread = column
    }
    out[(size_t)b * DIM + tid] = acc / sumw;
}

extern "C" void kernel_launch(void* const* d_in, const int* in_sizes, int n_in,
                              void* d_out, int out_size, void* d_ws, size_t ws_size,
                              hipStream_t stream)
{
    const float* query   = (const float*)d_in[0];   // [8, 2048, 256]
    const float* context = (const float*)d_in[1];   // [8, 4096, 256]
    float* out    = (float*)d_out;                  // [8, 1, 256]
    float* scores = (float*)d_ws;                   // [8, 4096] scratch

    dim3 grid1(LC / M_TILE, BATCH);                 // 32 x 8 blocks
    size_t shmem = (size_t)(M_TILE * LDP + 2 * B_BUF) * sizeof(float); // ~260 KB
    scores_wmma_kernel<<<grid1, 256, shmem, stream>>>(query, context, scores);

    softmax_ctxsum_kernel<<<dim3(BATCH), 256, 0, stream>>>(context, scores, out);
}